// ResGCNConv_39539468927576
// MI455X (gfx1250) — compile-verified
//
#include <hip/hip_runtime.h>

typedef float v2f __attribute__((ext_vector_type(2)));
typedef float v8f __attribute__((ext_vector_type(8)));

#define FEAT 64

// ---------------- elementwise helpers ----------------

__global__ void k_fill(float* __restrict__ p, float v, int n) {
  int i = blockIdx.x * blockDim.x + threadIdx.x;
  if (i < n) p[i] = v;
}

__global__ void k_deg(const int* __restrict__ dst, float* __restrict__ deg, int E) {
  int e = blockIdx.x * blockDim.x + threadIdx.x;
  if (e < E) atomicAdd(&deg[dst[e]], 1.0f);
}

__global__ void k_rsqrt(float* __restrict__ d, int n) {
  int i = blockIdx.x * blockDim.x + threadIdx.x;
  if (i < n) {
    float v = d[i];
    d[i] = (v > 0.0f) ? rsqrtf(v) : 0.0f;
  }
}

// ---------------- WMMA GEMM: Y[N,64] = X[N,64] @ W[64,64] ----------------
// One wave computes a 16-row x 64-col output slab using V_WMMA_F32_16X16X4_F32.
// A tile (16x4) lane layout (ISA 7.12.2): lanes 0-15 hold row M=lane, K=4s..4s+1
// in {v0,v1}; lanes 16-31 hold row M=lane-16, K=4s+2..4s+3. B tile (4x16)
// mirrors this with N = lane%16. D: VGPR r -> row r + 8*(lane>=16), col lane%16.
__global__ void k_gemm_wmma(const float* __restrict__ X, const float* __restrict__ W,
                            float* __restrict__ Y, int nrows) {
  __shared__ float Wt[FEAT * FEAT];  // transposed: Wt[n*64 + k] = W[k*64 + n]
  for (int i = threadIdx.x; i < FEAT * FEAT; i += blockDim.x) {
    int k = i >> 6, n = i & 63;
    Wt[n * FEAT + k] = W[i];
  }
  __syncthreads();

  const int lane = threadIdx.x & 31;
  const int wave = threadIdx.x >> 5;
  const int rowTile = blockIdx.x * (blockDim.x >> 5) + wave;
  const int row0 = rowTile * 16;
  if (row0 >= nrows) return;  // whole wave exits together; EXEC all-1 otherwise

  const int l2 = lane & 15;
  const int hi = lane >> 4;  // 0: K%4 in {0,1}; 1: K%4 in {2,3}

  // Preload all A fragments for this 16x64 x-tile (16 K-steps of 4).
  v2f a[16];
  const float* xrow = X + (size_t)(row0 + l2) * FEAT + 2 * hi;
#pragma unroll
  for (int s = 0; s < 16; ++s) a[s] = *(const v2f*)(xrow + 4 * s);

#pragma unroll
  for (int nt = 0; nt < 4; ++nt) {
    v8f acc = {};
    const float* wb = &Wt[(nt * 16 + l2) * FEAT + 2 * hi];
#pragma unroll
    for (int s = 0; s < 16; ++s) {
      v2f b = *(const v2f*)(wb + 4 * s);
      acc = __builtin_amdgcn_wmma_f32_16x16x4_f32(
          /*neg_a=*/false, a[s], /*neg_b=*/false, b,
          /*c_mod=*/(short)0, acc, /*reuse_a=*/false, /*reuse_b=*/false);
    }
    float* yout = Y + (size_t)row0 * FEAT + nt * 16 + l2;
    const int mbase = 8 * hi;
#pragma unroll
    for (int r = 0; r < 8; ++r) yout[(size_t)(r + mbase) * FEAT] = acc[r];
  }
}

// ---------------- edge scatter: out[dst] += dinv[src]*dinv[dst]*xw[src] ----------------
// One wave per edge; lane handles features lane and lane+32 (two coalesced 128B
// gather requests per edge, f32 global atomics land in L2 — all buffers are
// L2-resident at this problem size).
__global__ void k_scatter(const float* __restrict__ xw, const int* __restrict__ src,
                          const int* __restrict__ dst, const float* __restrict__ dinv,
                          float* __restrict__ out, int E) {
  const int lane = threadIdx.x & 31;
  const int e = (blockIdx.x * blockDim.x + threadIdx.x) >> 5;
  if (e >= E) return;
  const int s = __builtin_amdgcn_readfirstlane(src[e]);
  const int d = __builtin_amdgcn_readfirstlane(dst[e]);
  const float w = dinv[s] * dinv[d];
  const float v0 = xw[(size_t)s * FEAT + lane];
  const float v1 = xw[(size_t)s * FEAT + lane + 32];
  atomicAdd(&out[(size_t)d * FEAT + lane], w * v0);
  atomicAdd(&out[(size_t)d * FEAT + lane + 32], w * v1);
}

// h = relu(agg + dinv[i]^2 * xw + b)   (self-loop folded in; in-place on agg)
__global__ void k_post1(float* __restrict__ agg, const float* __restrict__ xw,
                        const float* __restrict__ dinv, const float* __restrict__ bias,
                        int n) {
  int idx = blockIdx.x * blockDim.x + threadIdx.x;
  if (idx >= n) return;
  int i = idx >> 6, f = idx & 63;
  float di = dinv[i];
  float v = agg[idx] + di * di * xw[idx] + bias[f];
  agg[idx] = fmaxf(v, 0.0f);
}

// out = relu(agg + dinv[i]^2 * xw + b + x)   (layer-2 + residual)
__global__ void k_post2(float* __restrict__ out, const float* __restrict__ xw,
                        const float* __restrict__ dinv, const float* __restrict__ bias,
                        const float* __restrict__ x, int n) {
  int idx = blockIdx.x * blockDim.x + threadIdx.x;
  if (idx >= n) return;
  int i = idx >> 6, f = idx & 63;
  float di = dinv[i];
  float v = out[idx] + di * di * xw[idx] + bias[f] + x[idx];
  out[idx] = fmaxf(v, 0.0f);
}

// ---------------- driver ----------------

extern "C" void kernel_launch(void* const* d_in, const int* in_sizes, int n_in,
                              void* d_out, int out_size, void* d_ws, size_t ws_size,
                              hipStream_t stream) {
  const float* x  = (const float*)d_in[0];
  const int*   ei = (const int*)d_in[1];
  const float* W1 = (const float*)d_in[2];
  const float* b1 = (const float*)d_in[3];
  const float* W2 = (const float*)d_in[4];
  const float* b2 = (const float*)d_in[5];

  const int N = in_sizes[0] / FEAT;
  const int E = in_sizes[1] / 2;
  const int* src = ei;       // edge_index[0]
  const int* dst = ei + E;   // edge_index[1]

  float* dinv = (float*)d_ws;                 // N
  float* bufA = dinv + N;                     // N*64 (xw scratch)
  float* bufB = bufA + (size_t)N * FEAT;      // N*64 (agg1 / h1)
  float* out  = (float*)d_out;

  const int NF = N * FEAT;
  const dim3 blk(256);
  const int gN   = (N + 255) / 256;
  const int gNF  = (NF + 255) / 256;
  const int gE   = (E + 255) / 256;
  const int gEw  = (E + 7) / 8;               // wave per edge, 8 waves/block
  const int rowTiles = (N + 15) / 16;
  const int gGemm = (rowTiles + 7) / 8;       // 8 waves (row-tiles) per block

  // degrees (self-loop => start at 1), zero layer-1 accumulator
  k_fill<<<gN, blk, 0, stream>>>(dinv, 1.0f, N);
  k_fill<<<gNF, blk, 0, stream>>>(bufB, 0.0f, NF);
  k_deg<<<gE, blk, 0, stream>>>(dst, dinv, E);
  k_rsqrt<<<gN, blk, 0, stream>>>(dinv, N);

  // layer 1
  k_gemm_wmma<<<gGemm, blk, 0, stream>>>(x, W1, bufA, N);
  k_scatter<<<gEw, blk, 0, stream>>>(bufA, src, dst, dinv, bufB, E);
  k_post1<<<gNF, blk, 0, stream>>>(bufB, bufA, dinv, b1, NF);

  // layer 2 (accumulate straight into d_out)
  k_gemm_wmma<<<gGemm, blk, 0, stream>>>(bufB, W2, bufA, N);
  k_fill<<<gNF, blk, 0, stream>>>(out, 0.0f, NF);
  k_scatter<<<gEw, blk, 0, stream>>>(bufA, src, dst, dinv, out, E);
  k_post2<<<gNF, blk, 0, stream>>>(out, bufA, dinv, b2, x, NF);
}